// ProductQuantizer_22686017258050
// MI455X (gfx1250) — compile-verified
//
#include <hip/hip_runtime.h>

typedef __attribute__((ext_vector_type(2))) float v2f;
typedef __attribute__((ext_vector_type(4))) float v4f;
typedef __attribute__((ext_vector_type(8))) float v8f;

#define PQ_N   16384
#define PQ_D   1024
#define PQ_M   64
#define PQ_K   256
#define PQ_SUB 16

// ---- prep: c2h[m*K+k] = 0.5 * ||codebooks[m][k]||^2  (1 thread per code)
__global__ void pq_c2half_kernel(const float* __restrict__ cb,
                                 float* __restrict__ c2h) {
    int i = blockIdx.x * blockDim.x + threadIdx.x;     // 0 .. M*K-1
    const float* p = cb + (size_t)i * PQ_SUB;
    float s = 0.f;
#pragma unroll
    for (int q = 0; q < PQ_SUB; ++q) s += p[q] * p[q];
    c2h[i] = 0.5f * s;
}

// ---- main: one wave = 16 embedding rows x 1 subspace.
// A (WMMA M axis) = 16 codes x 16 dims (K split into 4 chunks of 4)
// B (WMMA N axis) = -X^T : 16 dims x 16 embeddings (loop-invariant)
// C seed          = 0.5*||code||^2  -> D = 0.5c^2 - x.c (argmin-equivalent)
// Inner loop tracks only (tile_min, tile_id) via a v_min tree; the winning
// 16-code tile is recomputed once at the end to recover the exact index.
__global__ __launch_bounds__(256) void pq_wmma_kernel(
    const float* __restrict__ embeds, const float* __restrict__ cb,
    const float* __restrict__ c2h, float* __restrict__ out) {
    const int lane = threadIdx.x & 31;
    const int wib  = threadIdx.x >> 5;
    const int w    = blockIdx.x * 8 + wib;
    const int m    = w & (PQ_M - 1);
    const int tile = w >> 6;
    const int hi   = lane >> 4;       // half-wave select
    const int l16  = lane & 15;
    const int nrow = tile * 16 + l16; // embedding row owned on the N axis

    // B operand (negated X tile, transposed layout):
    // lane<16 holds dims {4kk, 4kk+1}, lane>=16 holds {4kk+2, 4kk+3}
    const float* xb = embeds + (size_t)nrow * PQ_D + m * PQ_SUB + hi * 2;
    v2f b[4];
#pragma unroll
    for (int kk = 0; kk < 4; ++kk) {
        v2f t = *(const v2f*)(xb + 4 * kk);
        b[kk] = -t;
    }

    const float* cbm = cb  + (size_t)m * PQ_K * PQ_SUB;
    const float* c2m = c2h + m * PQ_K;

    float best = 3.402823466e38f;
    int   bj   = 0;                   // winning 16-code tile (this half-wave)

    for (int j = 0; j < 16; ++j) {
        // seed accumulator with 0.5*c2 of the 8 codes this half-lane covers
        const float* c2p = c2m + j * 16 + hi * 8;
        v4f clo = *(const v4f*)(c2p);
        v4f chi = *(const v4f*)(c2p + 4);
        v8f acc;
        acc[0] = clo.x; acc[1] = clo.y; acc[2] = clo.z; acc[3] = clo.w;
        acc[4] = chi.x; acc[5] = chi.y; acc[6] = chi.z; acc[7] = chi.w;

        const float* ca = cbm + (size_t)(j * 16 + l16) * PQ_SUB + hi * 2;
#pragma unroll
        for (int kk = 0; kk < 4; ++kk) {
            v2f a = *(const v2f*)(ca + 4 * kk);
            acc = __builtin_amdgcn_wmma_f32_16x16x4_f32(
                false, a, false, b[kk], (short)0, acc, false, false);
        }

        // min tree over the 8 code rows (v_min3_num / dual v_min_num)
        float t0 = fminf(acc[0], acc[1]);
        float t1 = fminf(acc[2], acc[3]);
        float t2 = fminf(acc[4], acc[5]);
        float t3 = fminf(acc[6], acc[7]);
        float tm = fminf(fminf(t0, t1), fminf(t2, t3));
        if (tm < best) { best = tm; bj = j; }   // strict: keeps earliest tile
    }

    // merge the two half-wave candidates; key = j*2+hi is monotone in the
    // code-index base (j*16 + hi*8), preserving first-occurrence semantics
    int   mykey = bj * 2 + hi;
    float ob    = __shfl_xor(best, 16, 32);
    int   okey  = __shfl_xor(mykey, 16, 32);
    if (ob < best || (ob == best && okey < mykey)) { best = ob; mykey = okey; }
    const int jstar = mykey >> 1;
    const int hstar = mykey & 1;

    // recompute the winning tile (uniform control flow; EXEC all ones)
    {
        const float* c2p = c2m + jstar * 16 + hi * 8;
        v4f clo = *(const v4f*)(c2p);
        v4f chi = *(const v4f*)(c2p + 4);
        v8f acc;
        acc[0] = clo.x; acc[1] = clo.y; acc[2] = clo.z; acc[3] = clo.w;
        acc[4] = chi.x; acc[5] = chi.y; acc[6] = chi.z; acc[7] = chi.w;

        const float* ca = cbm + (size_t)(jstar * 16 + l16) * PQ_SUB + hi * 2;
#pragma unroll
        for (int kk = 0; kk < 4; ++kk) {
            v2f a = *(const v2f*)(ca + 4 * kk);
            acc = __builtin_amdgcn_wmma_f32_16x16x4_f32(
                false, a, false, b[kk], (short)0, acc, false, false);
        }

        // exact-match scan, downward so the lowest index wins on ties
        int idxl = 0x7fffffff;
#pragma unroll
        for (int r = 7; r >= 0; --r)
            if (acc[r] == best) idxl = jstar * 16 + hi * 8 + r;
        if (hi != hstar) idxl = 0x7fffffff;

        int oidx = __shfl_xor(idxl, 16, 32);
        int bidx = idxl < oidx ? idxl : oidx;

        // lanes 0..15: gather winning code vector (64B), write recon slice
        if (hi == 0) {
            const float* cv = cbm + (size_t)bidx * PQ_SUB;
            float*       op = out + (size_t)nrow * PQ_D + m * PQ_SUB;
#pragma unroll
            for (int q = 0; q < 4; ++q) ((v4f*)op)[q] = ((const v4f*)cv)[q];
        }
    }
}

extern "C" void kernel_launch(void* const* d_in, const int* in_sizes, int n_in,
                              void* d_out, int out_size, void* d_ws, size_t ws_size,
                              hipStream_t stream) {
    const float* embeds = (const float*)d_in[0];  // (16384, 1024) f32
    const float* cb     = (const float*)d_in[1];  // (64, 256, 16) f32
    float*       out    = (float*)d_out;          // (16384, 1024) f32
    float*       c2h    = (float*)d_ws;           // 64 KB scratch

    pq_c2half_kernel<<<(PQ_M * PQ_K) / 256, 256, 0, stream>>>(cb, c2h);

    const int waves  = (PQ_N / 16) * PQ_M;        // 65536 waves
    const int blocks = waves / 8;                 // 8 waves (256 thr) / block
    pq_wmma_kernel<<<blocks, 256, 0, stream>>>(embeds, cb, c2h, out);
}